// PainnModel_18451179503855
// MI455X (gfx1250) — compile-verified
//
#include <hip/hip_runtime.h>

// PaiNN GNN for MI455X (gfx1250). Bandwidth-bound: node arrays live in L2,
// edge-wise filter GEMM (K=20) fused into the message kernel to avoid a
// 245MB intermediate. All dense GEMMs use V_WMMA_F32_16X16X4_F32 (pure f32).

#define NNODES 10000
#define NEDGES 160000
#define HDIM   128
#define NL     3
#define EDIM   20
#define PI_F   3.14159265358979323846f
#define CUTOFF 5.0f

typedef __attribute__((ext_vector_type(2))) float v2f;
typedef __attribute__((ext_vector_type(8))) float v8f;

__device__ __forceinline__ float silu_f(float x) {
    return x / (1.0f + __expf(-x));
}

// ---------------------------------------------------------------------------
// C[M,N] = act(A[M,K] @ W[K,N] + bias[N])   (act: 0 = none, 1 = SiLU)
// grid.x = Ndim/16 (one B panel per block), grid.y = ceil(Mtiles/8)
// block = 256 threads = 8 waves; each wave owns one 16x16 C tile.
// B panel (K x 16) staged in LDS with padded stride to avoid bank conflicts.
// ---------------------------------------------------------------------------
__global__ __launch_bounds__(256) void wmma_gemm_bias(
    const float* __restrict__ A, const float* __restrict__ W,
    const float* __restrict__ bias, float* __restrict__ C,
    int Mdim, int Kdim, int Ndim, int act)
{
    __shared__ float ldsB[16 * 258];           // supports K <= 256
    const int KP   = Kdim + 2;                 // padded stride (floats)
    const int n0   = blockIdx.x * 16;
    const int wave = threadIdx.x >> 5;
    const int lane = threadIdx.x & 31;
    const int half = lane >> 4;                // 0: lanes 0-15, 1: lanes 16-31
    const int l15  = lane & 15;
    const int m0   = (blockIdx.y * 8 + wave) * 16;

    // Stage B panel: W[k][n0+n] -> ldsB[n*KP + k]  (coalesced 16-wide rows)
    for (int idx = threadIdx.x; idx < 16 * Kdim; idx += 256) {
        int n = idx & 15;
        int k = idx >> 4;
        ldsB[n * KP + k] = W[k * Ndim + n0 + n];
    }
    __syncthreads();

    if (m0 < Mdim) {                            // wave-uniform: EXEC all-1s
        v8f c = {0.f, 0.f, 0.f, 0.f, 0.f, 0.f, 0.f, 0.f};
        const float* Arow = A + (size_t)(m0 + l15) * Kdim;   // A: row per lane
        const float* Bcol = ldsB + l15 * KP;                 // B: col per lane
        for (int k0 = 0; k0 < Kdim; k0 += 4) {
            v2f a, b;
            const float* ap = Arow + k0 + half * 2;  // lanes 0-15: K{0,1}, 16-31: K{2,3}
            a.x = ap[0]; a.y = ap[1];
            const float* bp = Bcol + k0 + half * 2;  // same split on B's K rows
            b.x = bp[0]; b.y = bp[1];
            c = __builtin_amdgcn_wmma_f32_16x16x4_f32(
                    false, a, false, b, (short)0, c, false, false);
        }
        // C/D layout: VGPR j -> row m0 + half*8 + j, col n0 + l15
        float bn = bias[n0 + l15];
        float* Cp = C + (size_t)(m0 + half * 8) * Ndim + n0 + l15;
        #pragma unroll
        for (int j = 0; j < 8; ++j) {
            float x = c[j] + bn;
            if (act) x = silu_f(x);
            Cp[(size_t)j * Ndim] = x;
        }
    }
}

// ---------------------------------------------------------------------------
// Edge geometry: rbf (NE x 20), fcut (NE), unit (NE x 3)
// ---------------------------------------------------------------------------
__global__ void k_edge_geom(const float* __restrict__ diff,
                            const float* __restrict__ dist,
                            float* __restrict__ rbf,
                            float* __restrict__ fcut,
                            float* __restrict__ unit)
{
    int e = blockIdx.x * blockDim.x + threadIdx.x;
    if (e >= NEDGES) return;
    float d   = dist[e];
    float inv = 1.0f / d;
    #pragma unroll
    for (int j = 0; j < EDIM; ++j)
        rbf[e * EDIM + j] = sinf(d * (float)(j + 1) * (PI_F / CUTOFF)) * inv;
    fcut[e] = (d < CUTOFF) ? 0.5f * (cosf(PI_F * d / CUTOFF) + 1.0f) : 0.0f;
    unit[e * 3 + 0] = diff[e * 3 + 0] * inv;
    unit[e * 3 + 1] = diff[e * 3 + 1] * inv;
    unit[e * 3 + 2] = diff[e * 3 + 2] * inv;
}

// ---------------------------------------------------------------------------
// node_scalar = embed[z]; node_vector = 0   (launched over N*3*H threads)
// ---------------------------------------------------------------------------
__global__ void k_init_nodes(const int* __restrict__ z,
                             const float* __restrict__ embed,
                             float* __restrict__ ns,
                             float* __restrict__ nv)
{
    int i = blockIdx.x * blockDim.x + threadIdx.x;
    if (i < NNODES * 3 * HDIM) nv[i] = 0.0f;
    if (i < NNODES * HDIM) {
        int node = i >> 7, h = i & (HDIM - 1);
        ns[i] = embed[z[node] * HDIM + h];
    }
}

// ---------------------------------------------------------------------------
// Fused per-edge message kernel (one edge per 128-thread block).
// filter = (rbf @ Wf + bf) * fcut   computed on the fly (K=20),
// gathered against scalar_out[src] / nv_old[src], scattered via f32 atomics.
// ---------------------------------------------------------------------------
__global__ __launch_bounds__(128) void k_message(
    const int*   __restrict__ edge,
    const float* __restrict__ rbf,
    const float* __restrict__ fcut,
    const float* __restrict__ unit,
    const float* __restrict__ Wf,   // (20, 384)
    const float* __restrict__ bf,   // (384)
    const float* __restrict__ sout, // (N, 384)
    const float* __restrict__ nv_old,
    float* __restrict__ ns,
    float* __restrict__ nv_new)
{
    int e = blockIdx.x;
    int t = threadIdx.x;            // channel 0..127
    __shared__ float srbf[EDIM];
    __shared__ float su[3];
    __shared__ float sfc;
    if (t < EDIM) srbf[t] = rbf[e * EDIM + t];
    if (t < 3)    su[t]   = unit[e * 3 + t];
    if (t == 0)   sfc     = fcut[e];
    __syncthreads();

    int dst = edge[e * 2 + 0];
    int src = edge[e * 2 + 1];

    float f0 = bf[t], f1 = bf[HDIM + t], f2 = bf[2 * HDIM + t];
    #pragma unroll
    for (int k = 0; k < EDIM; ++k) {
        float r = srbf[k];
        const float* w = Wf + k * (3 * HDIM);
        f0 = fmaf(r, w[t], f0);
        f1 = fmaf(r, w[HDIM + t], f1);
        f2 = fmaf(r, w[2 * HDIM + t], f2);
    }
    float fc = sfc;
    const float* so = sout + (size_t)src * (3 * HDIM);
    float gsv = f0 * fc * so[t];                 // gate_sv
    float gev = f1 * fc * so[HDIM + t];          // gate_ev
    float ms  = f2 * fc * so[2 * HDIM + t];      // msg_s

    atomicAdd(&ns[(size_t)dst * HDIM + t], ms);

    const float* nvs = nv_old + (size_t)src * (3 * HDIM);
    float*       nvd = nv_new + (size_t)dst * (3 * HDIM);
    #pragma unroll
    for (int c = 0; c < 3; ++c) {
        float mv = nvs[c * HDIM + t] * gsv + gev * su[c];
        atomicAdd(&nvd[c * HDIM + t], mv);
    }
}

// ---------------------------------------------------------------------------
// mlp_in = concat(||Vv||_axis1 (N,H), ns (N,H)) -> (N, 2H)
// ---------------------------------------------------------------------------
__global__ void k_vnorm_concat(const float* __restrict__ Vv,
                               const float* __restrict__ ns,
                               float* __restrict__ mlp_in)
{
    int i = blockIdx.x * blockDim.x + threadIdx.x;
    if (i >= NNODES * HDIM) return;
    int node = i >> 7, h = i & (HDIM - 1);
    const float* v = Vv + (size_t)node * (3 * HDIM) + h;
    float v0 = v[0], v1 = v[HDIM], v2 = v[2 * HDIM];
    mlp_in[(size_t)node * (2 * HDIM) + h]        = sqrtf(v0 * v0 + v1 * v1 + v2 * v2);
    mlp_in[(size_t)node * (2 * HDIM) + HDIM + h] = ns[i];
}

// ---------------------------------------------------------------------------
// Apply update: nv += a_vv*Uv ;  ns += a_sv * <Uv,Vv>_axis1 + a_ss
// ---------------------------------------------------------------------------
__global__ void k_update(const float* __restrict__ mlp_out,
                         const float* __restrict__ Uv,
                         const float* __restrict__ Vv,
                         float* __restrict__ ns,
                         float* __restrict__ nv)
{
    int i = blockIdx.x * blockDim.x + threadIdx.x;
    if (i >= NNODES * HDIM) return;
    int node = i >> 7, h = i & (HDIM - 1);
    const float* mo = mlp_out + (size_t)node * (3 * HDIM);
    float a_vv = mo[h], a_sv = mo[HDIM + h], a_ss = mo[2 * HDIM + h];
    size_t base = (size_t)node * (3 * HDIM) + h;
    float dot = 0.0f;
    #pragma unroll
    for (int c = 0; c < 3; ++c) {
        float u = Uv[base + c * HDIM];
        nv[base + c * HDIM] += a_vv * u;
        dot = fmaf(u, Vv[base + c * HDIM], dot);
    }
    ns[i] += a_sv * dot + a_ss;
}

// ---------------------------------------------------------------------------
extern "C" void kernel_launch(void* const* d_in, const int* in_sizes, int n_in,
                              void* d_out, int out_size, void* d_ws, size_t ws_size,
                              hipStream_t stream)
{
    const int*   z     = (const int*)d_in[0];
    const int*   edge  = (const int*)d_in[1];
    const float* ediff = (const float*)d_in[2];
    const float* edist = (const float*)d_in[3];
    const float* embed = (const float*)d_in[4];
    const float* mfw   = (const float*)d_in[5];
    const float* mfb   = (const float*)d_in[6];
    const float* mw1   = (const float*)d_in[7];
    const float* mb1   = (const float*)d_in[8];
    const float* mw2   = (const float*)d_in[9];
    const float* mb2   = (const float*)d_in[10];
    const float* uUw   = (const float*)d_in[11];
    const float* uUb   = (const float*)d_in[12];
    const float* uVw   = (const float*)d_in[13];
    const float* uVb   = (const float*)d_in[14];
    const float* uw1   = (const float*)d_in[15];
    const float* ub1   = (const float*)d_in[16];
    const float* uw2   = (const float*)d_in[17];
    const float* ub2   = (const float*)d_in[18];
    const float* rw1   = (const float*)d_in[19];
    const float* rb1   = (const float*)d_in[20];
    const float* rw2   = (const float*)d_in[21];
    const float* rb2   = (const float*)d_in[22];

    float* ws = (float*)d_ws;
    size_t off = 0;
    float* ns   = ws + off; off += (size_t)NNODES * HDIM;       // node_scalar
    float* nvA  = ws + off; off += (size_t)NNODES * 3 * HDIM;   // node_vector A
    float* nvB  = ws + off; off += (size_t)NNODES * 3 * HDIM;   // node_vector B
    float* sout = ws + off; off += (size_t)NNODES * 3 * HDIM;   // scalar_out
    float* rbf  = ws + off; off += (size_t)NEDGES * EDIM;
    float* fcut = ws + off; off += (size_t)NEDGES;
    float* unit = ws + off; off += (size_t)NEDGES * 3;
    float* Uv   = ws + off; off += (size_t)NNODES * 3 * HDIM;
    float* Vv   = ws + off; off += (size_t)NNODES * 3 * HDIM;
    float* min_ = ws + off; off += (size_t)NNODES * 2 * HDIM;   // update MLP input
    float* tmp  = ws + off; off += (size_t)NNODES * HDIM;       // MLP hidden
    float* mout = ws + off; off += (size_t)NNODES * 3 * HDIM;   // update MLP out

    const int MT  = (NNODES / 16 + 7) / 8;        // 79  M-tile groups (N rows)
    const int MT3 = (3 * NNODES / 16 + 7) / 8;    // 235 M-tile groups (3N rows)

    k_edge_geom<<<(NEDGES + 255) / 256, 256, 0, stream>>>(ediff, edist, rbf, fcut, unit);
    k_init_nodes<<<(NNODES * 3 * HDIM + 255) / 256, 256, 0, stream>>>(z, embed, ns, nvA);

    float* nv_old = nvA;
    float* nv_new = nvB;
    for (int l = 0; l < NL; ++l) {
        // scalar_out = silu(ns @ w1 + b1) @ w2 + b2
        wmma_gemm_bias<<<dim3(8, MT), 256, 0, stream>>>(
            ns, mw1 + l * HDIM * HDIM, mb1 + l * HDIM, tmp, NNODES, HDIM, HDIM, 1);
        wmma_gemm_bias<<<dim3(24, MT), 256, 0, stream>>>(
            tmp, mw2 + l * HDIM * 384, mb2 + l * 384, sout, NNODES, HDIM, 384, 0);

        // messages read old nv; accumulate into a fresh copy
        hipMemcpyAsync(nv_new, nv_old, (size_t)NNODES * 3 * HDIM * sizeof(float),
                       hipMemcpyDeviceToDevice, stream);
        k_message<<<NEDGES, 128, 0, stream>>>(
            edge, rbf, fcut, unit, mfw + l * EDIM * 384, mfb + l * 384,
            sout, nv_old, ns, nv_new);

        // Uv / Vv projections over the (3N, H) flattening of node_vector
        wmma_gemm_bias<<<dim3(8, MT3), 256, 0, stream>>>(
            nv_new, uUw + l * HDIM * HDIM, uUb + l * HDIM, Uv, 3 * NNODES, HDIM, HDIM, 0);
        wmma_gemm_bias<<<dim3(8, MT3), 256, 0, stream>>>(
            nv_new, uVw + l * HDIM * HDIM, uVb + l * HDIM, Vv, 3 * NNODES, HDIM, HDIM, 0);

        k_vnorm_concat<<<(NNODES * HDIM + 255) / 256, 256, 0, stream>>>(Vv, ns, min_);

        wmma_gemm_bias<<<dim3(8, MT), 256, 0, stream>>>(
            min_, uw1 + l * 2 * HDIM * HDIM, ub1 + l * HDIM, tmp, NNODES, 2 * HDIM, HDIM, 1);
        wmma_gemm_bias<<<dim3(24, MT), 256, 0, stream>>>(
            tmp, uw2 + l * HDIM * 384, ub2 + l * 384, mout, NNODES, HDIM, 384, 0);

        k_update<<<(NNODES * HDIM + 255) / 256, 256, 0, stream>>>(mout, Uv, Vv, ns, nv_new);

        float* t2 = nv_old; nv_old = nv_new; nv_new = t2;
    }

    // readout
    wmma_gemm_bias<<<dim3(8, MT), 256, 0, stream>>>(ns, rw1, rb1, tmp, NNODES, HDIM, HDIM, 1);
    wmma_gemm_bias<<<dim3(8, MT), 256, 0, stream>>>(tmp, rw2, rb2, (float*)d_out,
                                                    NNODES, HDIM, HDIM, 0);
}